// DiffKS_15625091023331
// MI455X (gfx1250) — compile-verified
//
#include <hip/hip_runtime.h>

// CDNA5 / gfx1250: wave32, WMMA f32 16x16x4 (full-precision matrix pipe).
typedef __attribute__((ext_vector_type(2))) float v2f;
typedef __attribute__((ext_vector_type(8))) float v8f;

#define NB     4        // batch
#define BT     64000    // time steps
#define BN     588      // filter taps (multiple of 4)
#define TILE_T 128      // t-rows per block
#define NWAVES 8        // 256 threads = 8 wave32

// x[b,t] = y[b,t] + sum_k A[b,t,k] * ypad[b, t + BN-1 - k]
// ypad[j] = zi[b, BN-1-j] (j<BN) else y[b, j-BN]
__global__ __launch_bounds__(256) void diffks_fir_wmma(
    const float* __restrict__ y,
    const float* __restrict__ A,
    const float* __restrict__ zi,
    float* __restrict__ x)
{
    __shared__ float ypad[TILE_T + BN];          // 716 floats: window for this block
    __shared__ float diag[NWAVES][32 * 8];       // accumulator spill for diagonal pick

    const int tid   = threadIdx.x;
    const int lane  = tid & 31;
    const int wave  = tid >> 5;
    const int tiles = BT / TILE_T;               // 500
    const int b     = blockIdx.x / tiles;
    const int t0b   = (blockIdx.x % tiles) * TILE_T;

    const float* yb  = y  + (size_t)b * BT;
    const float* zib = zi + (size_t)b * BN;

    // ---- stage the shared ypad window: ypad[u] = ypad_global(t0b + u) ----
    for (int u = tid; u < TILE_T + BN - 1; u += 256) {
        const int j = t0b + u;
        ypad[u] = (j < BN) ? zib[BN - 1 - j] : yb[j - BN];
    }
    __syncthreads();

    // ---- per-wave 16-row tile ----
    const int t0   = t0b + wave * 16;            // absolute first row of this wave
    const int n    = lane & 15;                  // matrix row/col owned by this lane
    const int qb   = (lane < 16) ? 0 : 2;        // K sub-pair per WMMA fragment layout
    const int base = wave * 16 + (BN - 1);       // ypad-local offset

    // A fragment source: lane covers row (t0+n), columns (k0+qb, k0+qb+1)
    const float* arow = A + ((size_t)b * BT + (size_t)(t0 + n)) * BN + qb;

    v8f acc = {0.f, 0.f, 0.f, 0.f, 0.f, 0.f, 0.f, 0.f};

    #pragma unroll 4
    for (int k0 = 0; k0 < BN; k0 += 4) {
        // A-matrix 16x4 f32 fragment (ISA 7.12.2): lanes0-15 K={0,1}, lanes16-31 K={2,3}
        const v2f afrag = *reinterpret_cast<const v2f*>(arow + k0);  // 8B aligned

        // B-matrix 4x16: B[q, n] = shifts[t0+n, k0+q] = ypad[base + n - k0 - q]
        const int bi = base + n - k0 - qb;
        v2f bfrag;
        bfrag.x = ypad[bi];          // K = qb
        bfrag.y = ypad[bi - 1];      // K = qb + 1

        // D = A x B + C, f32 accumulate; diagonal D[i,i] is the wanted dot product
        acc = __builtin_amdgcn_wmma_f32_16x16x4_f32(
            /*neg_a=*/false, afrag,
            /*neg_b=*/false, bfrag,
            /*c_mod=*/(short)0, acc,
            /*reuse_a=*/false, /*reuse_b=*/false);
    }

    // ---- extract diagonal: (i,i) lives at VGPR i / lane i  (i<8)
    //                        and VGPR i-8 / lane i+16        (i>=8) ----
    float* dg = diag[wave];
    #pragma unroll
    for (int r = 0; r < 8; ++r) dg[lane * 8 + r] = acc[r];
    __syncthreads();

    if (lane < 16) {
        const int i = lane;
        const float v = (i < 8) ? dg[i * 8 + i] : dg[(i + 16) * 8 + (i - 8)];
        x[(size_t)b * BT + t0 + i] = yb[t0 + i] + v;
    }
}

extern "C" void kernel_launch(void* const* d_in, const int* in_sizes, int n_in,
                              void* d_out, int out_size, void* d_ws, size_t ws_size,
                              hipStream_t stream) {
    const float* y  = (const float*)d_in[0];   // (B, T)
    const float* A  = (const float*)d_in[1];   // (B, T, N)
    const float* zi = (const float*)d_in[2];   // (B, N)
    float* x = (float*)d_out;                  // (B, T)

    const int grid = NB * (BT / TILE_T);       // 4 * 500 = 2000 blocks
    diffks_fir_wmma<<<grid, 256, 0, stream>>>(y, A, zi, x);
}